// MultiHeadAttention_7945689497895
// MI455X (gfx1250) — compile-verified
//
#include <hip/hip_runtime.h>
#include <hip/hip_bf16.h>

// MI455X / gfx1250, wave32. Reference collapses to linear attention:
//   out_head = (scale*mt * Q @ (K^T diag(m) V)) / (scale*mt * Q.(K^T m) + EPS)
// All matmuls in fp32 via V_WMMA_F32_16X16X4_F32 (only fp32 WMMA on CDNA5).

typedef __attribute__((ext_vector_type(2))) float v2f;
typedef __attribute__((ext_vector_type(8))) float v8f;

#define SEQ_T   1024
#define DMODEL  768
#define NHEADS  12
#define UHEAD   64
#define EPSF    1e-7f

__device__ __forceinline__ v8f wmma_f32(v2f a, v2f b, v8f c) {
  // (neg_a, A, neg_b, B, c_mod, C, reuse_a, reuse_b)
  return __builtin_amdgcn_wmma_f32_16x16x4_f32(false, a, false, b, (short)0, c,
                                               false, false);
}

// ---------------------------------------------------------------------------
// C[M,N] = A[M,K] @ B[K,N] (+ optional residual R[M,N]).
// Requires M%32==0, N%64==0, K%32==0. Block = 256 thr = 8 waves (2x4 WMMA tiles).
// ---------------------------------------------------------------------------
__global__ __launch_bounds__(256) void gemm_wmma_f32_kernel(
    const float* __restrict__ A, const float* __restrict__ B,
    const float* __restrict__ R, float* __restrict__ C,
    int M, int N, int K) {
  __shared__ float As[32][33];
  __shared__ float Bs[32][65];
  const int tid  = threadIdx.x;
  const int lane = tid & 31, wid = tid >> 5;
  const int half = lane >> 4, l = lane & 15;
  const int wm = (wid & 1) * 16, wn = (wid >> 1) * 16;
  const int m0 = blockIdx.x * 32, n0 = blockIdx.y * 64;
  v8f acc = {};
  for (int k0 = 0; k0 < K; k0 += 32) {
    {  // A tile 32x32 : one float4 per thread
      int row = tid >> 3, c4 = (tid & 7) * 4;
      float4 va = *(const float4*)(A + (size_t)(m0 + row) * K + k0 + c4);
      As[row][c4 + 0] = va.x; As[row][c4 + 1] = va.y;
      As[row][c4 + 2] = va.z; As[row][c4 + 3] = va.w;
    }
#pragma unroll
    for (int j = 0; j < 2; ++j) {  // B tile 32x64 : two float4 per thread
      int e = tid + j * 256;
      int row = e >> 4, c4 = (e & 15) * 4;
      float4 vb = *(const float4*)(B + (size_t)(k0 + row) * N + n0 + c4);
      Bs[row][c4 + 0] = vb.x; Bs[row][c4 + 1] = vb.y;
      Bs[row][c4 + 2] = vb.z; Bs[row][c4 + 3] = vb.w;
    }
    __syncthreads();
#pragma unroll
    for (int kk = 0; kk < 32; kk += 4) {
      v2f a, b;
      // 32-bit A 16x4 layout: V0 = K{0,2}, V1 = K{1,3}, split at lane 16
      a.x = As[wm + l][kk + half * 2 + 0];
      a.y = As[wm + l][kk + half * 2 + 1];
      // 32-bit B 4x16 layout: V0 = rows K{0,2}, V1 = rows K{1,3}
      b.x = Bs[kk + half * 2 + 0][wn + l];
      b.y = Bs[kk + half * 2 + 1][wn + l];
      acc = wmma_f32(a, b, acc);
    }
    __syncthreads();
  }
#pragma unroll
  for (int i = 0; i < 8; ++i) {  // C/D: VGPR i -> rows {i, 8+i} by lane half
    int row = m0 + wm + half * 8 + i;
    int col = n0 + wn + l;
    float v = acc[i];
    if (R) v += R[(size_t)row * N + col];
    C[(size_t)row * N + col] = v;
  }
}

// ---------------------------------------------------------------------------
// Per (b,h): Mh[64][64] = sum_s m_s * K[s]^T V[s];  Ks[64] = sum_s m_s * K[s].
// One 512-thread block (16 waves, 4x4 WMMA tiles over 64x64).
// ---------------------------------------------------------------------------
__global__ __launch_bounds__(512) void kv_outer_kernel(
    const float* __restrict__ Kmat, const float* __restrict__ Vmat,
    const unsigned char* __restrict__ mask,
    float* __restrict__ Mh, float* __restrict__ Ksum) {
  __shared__ float Kt[32][65];
  __shared__ float Vt[32][65];
  __shared__ float red[8][64];
  const int bh = blockIdx.x;
  const int b = bh / NHEADS, h = bh % NHEADS;
  const int tid  = threadIdx.x;
  const int lane = tid & 31, wid = tid >> 5;
  const int half = lane >> 4, l = lane & 15;
  const int u0 = (wid & 3) * 16, w0 = (wid >> 2) * 16;
  const float* Kb = Kmat + (size_t)b * SEQ_T * DMODEL + h * UHEAD;
  const float* Vb = Vmat + (size_t)b * SEQ_T * DMODEL + h * UHEAD;
  const unsigned char* mb = mask + (size_t)b * SEQ_T;
  const int ru = tid & 63, rg = tid >> 6;
  v8f acc = {};
  float kpart = 0.f;
  for (int s0 = 0; s0 < SEQ_T; s0 += 32) {
    {  // stage 32x64 of (m*K) and V : one float4 per matrix per thread
      int row = tid >> 4, c4 = (tid & 15) * 4;
      float m = mb[s0 + row] ? 1.f : 0.f;
      float4 vk = *(const float4*)(Kb + (size_t)(s0 + row) * DMODEL + c4);
      Kt[row][c4 + 0] = vk.x * m; Kt[row][c4 + 1] = vk.y * m;
      Kt[row][c4 + 2] = vk.z * m; Kt[row][c4 + 3] = vk.w * m;
      float4 vv = *(const float4*)(Vb + (size_t)(s0 + row) * DMODEL + c4);
      Vt[row][c4 + 0] = vv.x; Vt[row][c4 + 1] = vv.y;
      Vt[row][c4 + 2] = vv.z; Vt[row][c4 + 3] = vv.w;
    }
    __syncthreads();
#pragma unroll
    for (int kk = 0; kk < 32; kk += 4) {
      v2f a, bf;
      // A = (m*K)^T : A[u][s] = Kt[s][u]  (contiguous-lane LDS reads)
      a.x = Kt[kk + half * 2 + 0][u0 + l];
      a.y = Kt[kk + half * 2 + 1][u0 + l];
      bf.x = Vt[kk + half * 2 + 0][w0 + l];
      bf.y = Vt[kk + half * 2 + 1][w0 + l];
      acc = wmma_f32(a, bf, acc);
    }
#pragma unroll
    for (int r = 0; r < 4; ++r) kpart += Kt[rg * 4 + r][ru];
    __syncthreads();
  }
#pragma unroll
  for (int i = 0; i < 8; ++i) {
    int row = u0 + half * 8 + i, col = w0 + l;
    Mh[(size_t)bh * 4096 + row * 64 + col] = acc[i];
  }
  red[rg][ru] = kpart;
  __syncthreads();
  if (tid < 64) {
    float s = 0.f;
#pragma unroll
    for (int g = 0; g < 8; ++g) s += red[g][tid];
    Ksum[(size_t)bh * 64 + tid] = s;
  }
}

// ---------------------------------------------------------------------------
// Per (b,h, 32-row tile of T): Out = (scale*mt * Q@Mh) / (scale*mt * Q.ksum + EPS)
// Block = 256 thr = 8 waves (2x4 tiles over 32x64).
// ---------------------------------------------------------------------------
__global__ __launch_bounds__(256) void attn_apply_kernel(
    const float* __restrict__ Q, const float* __restrict__ Mh,
    const float* __restrict__ Ksum, const unsigned char* __restrict__ mask,
    float* __restrict__ Out, float scale) {
  __shared__ float Qs[32][65];
  __shared__ float Ms[64][65];
  __shared__ float ksum_s[64];
  __shared__ float redd[32][8];
  __shared__ float rowdot[32];
  const int t0 = blockIdx.x * 32;
  const int bh = blockIdx.y;
  const int b = bh / NHEADS, h = bh % NHEADS;
  const int tid  = threadIdx.x;
  const int lane = tid & 31, wid = tid >> 5;
  const int half = lane >> 4, l = lane & 15;
  const int wm = (wid & 1) * 16, wn = (wid >> 1) * 16;
  const float* Qb = Q + (size_t)b * SEQ_T * DMODEL + h * UHEAD;
#pragma unroll
  for (int j = 0; j < 2; ++j) {  // Q tile 32x64
    int e = tid + j * 256;
    int row = e >> 4, c4 = (e & 15) * 4;
    float4 v = *(const float4*)(Qb + (size_t)(t0 + row) * DMODEL + c4);
    Qs[row][c4 + 0] = v.x; Qs[row][c4 + 1] = v.y;
    Qs[row][c4 + 2] = v.z; Qs[row][c4 + 3] = v.w;
  }
#pragma unroll
  for (int j = 0; j < 4; ++j) {  // Mh 64x64
    int e = tid + j * 256;
    int row = e >> 4, c4 = (e & 15) * 4;
    float4 v = *(const float4*)(Mh + (size_t)bh * 4096 + row * 64 + c4);
    Ms[row][c4 + 0] = v.x; Ms[row][c4 + 1] = v.y;
    Ms[row][c4 + 2] = v.z; Ms[row][c4 + 3] = v.w;
  }
  if (tid < 64) ksum_s[tid] = Ksum[(size_t)bh * 64 + tid];
  __syncthreads();
  {  // rowdot[t] = Q[t] . ksum  (8 segments of 8 per row)
    int row = tid >> 3, seg = tid & 7;
    float p = 0.f;
#pragma unroll
    for (int u = 0; u < 8; ++u) p += Qs[row][seg * 8 + u] * ksum_s[seg * 8 + u];
    redd[row][seg] = p;
  }
  __syncthreads();
  if (tid < 32) {
    float s = 0.f;
#pragma unroll
    for (int g = 0; g < 8; ++g) s += redd[tid][g];
    rowdot[tid] = s;
  }
  __syncthreads();
  v8f acc = {};
#pragma unroll
  for (int kk = 0; kk < 64; kk += 4) {
    v2f a, bf;
    a.x = Qs[wm + l][kk + half * 2 + 0];
    a.y = Qs[wm + l][kk + half * 2 + 1];
    bf.x = Ms[kk + half * 2 + 0][wn + l];
    bf.y = Ms[kk + half * 2 + 1][wn + l];
    acc = wmma_f32(a, bf, acc);
  }
  const unsigned char* mb = mask + (size_t)b * SEQ_T;
#pragma unroll
  for (int i = 0; i < 8; ++i) {
    int rl = wm + half * 8 + i;
    int t = t0 + rl;
    float mt = mb[t] ? 1.f : 0.f;
    float denom = scale * mt * rowdot[rl] + EPSF;
    float val = (scale * mt * acc[i]) / denom;
    Out[((size_t)b * SEQ_T + t) * DMODEL + h * UHEAD + wn + l] = val;
  }
}

// ---------------------------------------------------------------------------
// In-place row "norm" (faithful source bug: std = mean). Wave per row.
// ---------------------------------------------------------------------------
__global__ __launch_bounds__(256) void rownorm_kernel(
    float* __restrict__ Y, const float* __restrict__ gamma,
    const float* __restrict__ beta) {
  const int tid  = threadIdx.x;
  const int lane = tid & 31, wid = tid >> 5;
  const int row  = blockIdx.x * 8 + wid;
  float* yr = Y + (size_t)row * DMODEL;
  float v[24];
  float s = 0.f;
#pragma unroll
  for (int j = 0; j < 24; ++j) { v[j] = yr[lane + j * 32]; s += v[j]; }
#pragma unroll
  for (int off = 16; off > 0; off >>= 1) s += __shfl_xor(s, off, 32);
  float mean = s * (1.f / (float)DMODEL);
  float inv  = 1.f / (mean + EPSF);
#pragma unroll
  for (int j = 0; j < 24; ++j) {
    int c = lane + j * 32;
    yr[c] = gamma[c] * (v[j] - mean) * inv + beta[c];
  }
}

// ---------------------------------------------------------------------------
extern "C" void kernel_launch(void* const* d_in, const int* in_sizes, int n_in,
                              void* d_out, int out_size, void* d_ws,
                              size_t ws_size, hipStream_t stream) {
  (void)in_sizes; (void)n_in; (void)out_size; (void)ws_size;
  const float*         x       = (const float*)d_in[0];
  const unsigned char* mask    = (const unsigned char*)d_in[1];  // numpy bool
  const float*         w_q     = (const float*)d_in[2];
  const float*         w_k     = (const float*)d_in[3];
  const float*         w_v     = (const float*)d_in[4];
  const float*         w_final = (const float*)d_in[5];
  const float*         gamma   = (const float*)d_in[6];
  const float*         beta    = (const float*)d_in[7];
  float* out = (float*)d_out;

  const size_t MT = 8 * (size_t)SEQ_T;  // 8192 rows
  float* q    = (float*)d_ws;
  float* k    = q    + MT * DMODEL;
  float* v    = k    + MT * DMODEL;
  float* attn = v    + MT * DMODEL;
  float* Mh   = attn + MT * DMODEL;
  float* Ks   = Mh   + (size_t)8 * NHEADS * 64 * 64;
  // total ws use: 4*25.2MB + ~1.6MB ~= 102 MB

  const float scale = 0.036084391824351615f;  // 768^-0.5

  dim3 gg(256, 12), bb(256);
  gemm_wmma_f32_kernel<<<gg, bb, 0, stream>>>(x, w_q, nullptr, q, 8192, 768, 768);
  gemm_wmma_f32_kernel<<<gg, bb, 0, stream>>>(x, w_k, nullptr, k, 8192, 768, 768);
  gemm_wmma_f32_kernel<<<gg, bb, 0, stream>>>(x, w_v, nullptr, v, 8192, 768, 768);
  kv_outer_kernel<<<8 * NHEADS, 512, 0, stream>>>(k, v, mask, Mh, Ks);
  attn_apply_kernel<<<dim3(SEQ_T / 32, 8 * NHEADS), 256, 0, stream>>>(
      q, Mh, Ks, mask, attn, scale);
  gemm_wmma_f32_kernel<<<gg, bb, 0, stream>>>(attn, w_final, x, out, 8192, 768, 768);
  rownorm_kernel<<<8192 / 8, 256, 0, stream>>>(out, gamma, beta);
}